// AttentionBlock_7035156430892
// MI455X (gfx1250) — compile-verified
//
#include <hip/hip_runtime.h>

// ---------------------------------------------------------------------------
// AttentionBlock forward for MI455X (gfx1250), fp32 end-to-end.
// pproc(x,W,b) == x@W.T + b in forward (zp * stopgrad(z/zp) == z to ~1 ulp).
// Matrix math on the CDNA5 matrix pipe via V_WMMA_F32_16X16X4_F32.
// Tile staging via GLOBAL_LOAD_ASYNC_TO_LDS_B128 (ASYNCcnt) with an XOR-swizzled
// LDS layout (16B-aligned lane writes, bank-conflict-free b64 fragment reads).
// ---------------------------------------------------------------------------

#define BATCH  4
#define S_LEN  2048
#define NHEAD  16
#define HDIM   64
#define HID    1024

typedef float v2f __attribute__((ext_vector_type(2)));
typedef float v8f __attribute__((ext_vector_type(8)));

__device__ __forceinline__ v8f wmma4(v2f a, v2f b, v8f c) {
  // D = A(16x4) x B(4x16) + C(16x16), full fp32
  return __builtin_amdgcn_wmma_f32_16x16x4_f32(false, a, false, b, (short)0, c,
                                               false, false);
}

// Async global->LDS copy, 16B per lane, GV addressing (64-bit vaddr pair).
__device__ __forceinline__ void async_copy_b128(unsigned lds_byte_off,
                                                const float* gptr) {
  asm volatile("global_load_async_to_lds_b128 %0, %1, off"
               :
               : "v"(lds_byte_off), "v"(gptr));
}
__device__ __forceinline__ void async_wait0() {
  asm volatile("s_wait_asynccnt 0x0" ::: "memory");
}
// Workgroup-relative LDS byte offset (low 32 bits of generic shared pointer).
__device__ __forceinline__ unsigned lds_off(const void* p) {
  return (unsigned)(unsigned long long)p;
}
// XOR swizzle: 16B chunk c of row r -> byte offset within a 128x16f tile.
__device__ __forceinline__ unsigned swz(int r, int c) {
  return (unsigned)(r * 64 + ((c ^ ((r >> 2) & 3)) << 4));
}

// Y[m,n] = sum_k A[m,k]*B[n,k] (+bias[n]) (+res[m,n]); A: MxK lda=K, B: NxK ldb=K.
// Macro tile 128x128, BK=16, 8 waves of 32x64. Batched over grid.z via strides.
template <bool HEAD_MAJOR_OUT, bool ADD_RES, bool HAS_BIAS>
__global__ __launch_bounds__(256) void gemm_bt_kernel(
    const float* __restrict__ A, const float* __restrict__ B,
    const float* __restrict__ bias, const float* __restrict__ res,
    float* __restrict__ Y, int K, int ldy,
    long long aBatch, long long bBatch, long long yBatch) {
  __shared__ float sA[2][128 * 16];
  __shared__ float sB[2][128 * 16];

  const int tid  = threadIdx.x;
  const int lane = tid & 31;
  const int wave = tid >> 5;
  const int row0 = blockIdx.y * 128;
  const int col0 = blockIdx.x * 128;
  A += (long long)blockIdx.z * aBatch;
  B += (long long)blockIdx.z * bBatch;
  Y += (long long)blockIdx.z * yBatch;

  const int rw0 = (wave >> 1) * 32;  // wave tile: 32 rows
  const int cw0 = (wave & 1) * 64;   //            64 cols

  // Tile fill: 512 16B chunks per 128x16 tile; each thread owns 2 per matrix.
  const int fr0 = tid >> 2, fc = tid & 3;
  const int fr1 = fr0 + 64;
  const unsigned la0 = lds_off(&sA[0][0]) + swz(fr0, fc);
  const unsigned la1 = lds_off(&sA[0][0]) + swz(fr1, fc);
  const unsigned lb0 = lds_off(&sB[0][0]) + swz(fr0, fc);
  const unsigned lb1 = lds_off(&sB[0][0]) + swz(fr1, fc);
  const float* gA0 = A + (long long)(row0 + fr0) * K + fc * 4;
  const float* gA1 = A + (long long)(row0 + fr1) * K + fc * 4;
  const float* gB0 = B + (long long)(col0 + fr0) * K + fc * 4;
  const float* gB1 = B + (long long)(col0 + fr1) * K + fc * 4;

  auto issue = [&](int kt, int buf) {
    const int k0 = kt * 16;
    const unsigned bo = (unsigned)buf * (128u * 16u * 4u);
    async_copy_b128(la0 + bo, gA0 + k0);
    async_copy_b128(la1 + bo, gA1 + k0);
    async_copy_b128(lb0 + bo, gB0 + k0);
    async_copy_b128(lb1 + bo, gB1 + k0);
  };

  // Fragment addressing (loop-invariant parts)
  int rA[2]; int xA[2];
#pragma unroll
  for (int i = 0; i < 2; ++i) {
    rA[i] = rw0 + i * 16 + (lane & 15);
    xA[i] = (rA[i] >> 2) & 3;
  }
  int cB[4]; int xB[4];
#pragma unroll
  for (int j = 0; j < 4; ++j) {
    cB[j] = cw0 + j * 16 + (lane & 15);
    xB[j] = (cB[j] >> 2) & 3;
  }
  const int hl2 = (lane >> 4) << 1;  // lanes 0-15: K+0..1, lanes 16-31: K+2..3

  v8f acc[2][4];
#pragma unroll
  for (int i = 0; i < 2; ++i)
#pragma unroll
    for (int j = 0; j < 4; ++j)
#pragma unroll
      for (int e = 0; e < 8; ++e) acc[i][j][e] = 0.0f;

  const int nk = K >> 4;
  issue(0, 0);
  async_wait0();
  __syncthreads();
  for (int kt = 0; kt < nk; ++kt) {
    const int cur = kt & 1;
    if (kt + 1 < nk) {
      issue(kt + 1, cur ^ 1);
      if (kt + 2 < nk) {
        __builtin_prefetch(gA0 + (kt + 2) * 16, 0, 1);
        __builtin_prefetch(gB0 + (kt + 2) * 16, 0, 1);
      }
    }
    const float* a = sA[cur];
    const float* b = sB[cur];
#pragma unroll
    for (int ks = 0; ks < 16; ks += 4) {
      const int ch = ks >> 2;  // compile-time chunk index
      v2f af[2], bf[4];
#pragma unroll
      for (int i = 0; i < 2; ++i) {
        float2 t = *(const float2*)&a[rA[i] * 16 + ((ch ^ xA[i]) << 2) + hl2];
        af[i].x = t.x; af[i].y = t.y;
      }
#pragma unroll
      for (int j = 0; j < 4; ++j) {
        float2 t = *(const float2*)&b[cB[j] * 16 + ((ch ^ xB[j]) << 2) + hl2];
        bf[j].x = t.x; bf[j].y = t.y;
      }
#pragma unroll
      for (int i = 0; i < 2; ++i)
#pragma unroll
        for (int j = 0; j < 4; ++j) acc[i][j] = wmma4(af[i], bf[j], acc[i][j]);
    }
    async_wait0();   // next-buffer copies landed (own wave)
    __syncthreads(); // whole workgroup: writes visible, cur-buffer reads done
  }

  // Epilogue: C/D layout -> lanes 0-15 hold M=r, lanes 16-31 hold M=8+r
#pragma unroll
  for (int i = 0; i < 2; ++i) {
#pragma unroll
    for (int j = 0; j < 4; ++j) {
      const int mBase = row0 + rw0 + i * 16 + ((lane >> 4) << 3);
      const int n = col0 + cw0 + j * 16 + (lane & 15);
      const float bv = HAS_BIAS ? bias[n] : 0.0f;
#pragma unroll
      for (int e = 0; e < 8; ++e) {
        const int m = mBase + e;
        float val = acc[i][j][e] + bv;
        if (ADD_RES) val += res[(long long)m * ldy + n];
        if (HEAD_MAJOR_OUT) {
          const int bb = m >> 11, s = m & (S_LEN - 1);
          const int h = n >> 6, d = n & (HDIM - 1);
          Y[(((long long)(bb * NHEAD + h) << 11) + s) * HDIM + d] = val;
        } else {
          Y[(long long)m * ldy + n] = val;
        }
      }
    }
  }
}

// ctx[q,d] = sum_k probs[q,k] * V[k,d] per (b,h). M=2048, N=64, K=2048.
__global__ __launch_bounds__(256) void ctx_gemm_kernel(
    const float* __restrict__ P, const float* __restrict__ V,
    float* __restrict__ C) {
  __shared__ float sP[2][128 * 16];
  __shared__ float sV[2][16 * 64];

  const int tid = threadIdx.x, lane = tid & 31, wave = tid >> 5;
  const int row0 = blockIdx.x * 128;
  const int z = blockIdx.y;  // b*NHEAD + h
  const int bb = z >> 4, h = z & 15;
  const float* Pz = P + (long long)z * S_LEN * S_LEN;
  const float* Vz = V + (long long)z * S_LEN * HDIM;
  float* Cz = C + (long long)bb * S_LEN * HID + h * HDIM;

  const int rw0 = (wave >> 1) * 32;  // 4 waves over M
  const int cw0 = (wave & 1) * 32;   // 2 waves over N=64

  // P tile: 512 chunks -> 2/thread (swizzled). V tile: 256 chunks -> 1/thread.
  const int fr0 = tid >> 2, fc = tid & 3;
  const int fr1 = fr0 + 64;
  const unsigned lp0 = lds_off(&sP[0][0]) + swz(fr0, fc);
  const unsigned lp1 = lds_off(&sP[0][0]) + swz(fr1, fc);
  const int vr = tid >> 4, vc = tid & 15;
  const unsigned lv = lds_off(&sV[0][0]) + (unsigned)(vr * 256 + vc * 16);
  const float* gP0 = Pz + (long long)(row0 + fr0) * S_LEN + fc * 4;
  const float* gP1 = Pz + (long long)(row0 + fr1) * S_LEN + fc * 4;
  const float* gV  = Vz + (long long)vr * HDIM + vc * 4;

  auto issue = [&](int kt, int buf) {
    const int k0 = kt * 16;
    async_copy_b128(lp0 + (unsigned)buf * 8192u, gP0 + k0);
    async_copy_b128(lp1 + (unsigned)buf * 8192u, gP1 + k0);
    async_copy_b128(lv + (unsigned)buf * 4096u, gV + (long long)k0 * HDIM);
  };

  int rA[2]; int xA[2];
#pragma unroll
  for (int i = 0; i < 2; ++i) {
    rA[i] = rw0 + i * 16 + (lane & 15);
    xA[i] = (rA[i] >> 2) & 3;
  }
  const int hl2 = (lane >> 4) << 1;

  v8f acc[2][2];
#pragma unroll
  for (int i = 0; i < 2; ++i)
#pragma unroll
    for (int j = 0; j < 2; ++j)
#pragma unroll
      for (int e = 0; e < 8; ++e) acc[i][j][e] = 0.0f;

  const int nk = S_LEN / 16;  // 128
  issue(0, 0);
  async_wait0();
  __syncthreads();
  for (int kt = 0; kt < nk; ++kt) {
    const int cur = kt & 1;
    if (kt + 1 < nk) {
      issue(kt + 1, cur ^ 1);
      if (kt + 2 < nk) __builtin_prefetch(gP0 + (kt + 2) * 16, 0, 1);
    }
    const float* p = sP[cur];
    const float* v = sV[cur];
#pragma unroll
    for (int ks = 0; ks < 16; ks += 4) {
      const int ch = ks >> 2;
      const int kk = ks + hl2;
      v2f af[2], bf[2];
#pragma unroll
      for (int i = 0; i < 2; ++i) {
        float2 t = *(const float2*)&p[rA[i] * 16 + ((ch ^ xA[i]) << 2) + hl2];
        af[i].x = t.x; af[i].y = t.y;
      }
#pragma unroll
      for (int j = 0; j < 2; ++j) {  // V is [K,N]-major: strided b32 pair
        const int n = cw0 + j * 16 + (lane & 15);
        bf[j].x = v[kk * HDIM + n];
        bf[j].y = v[(kk + 1) * HDIM + n];
      }
#pragma unroll
      for (int i = 0; i < 2; ++i)
#pragma unroll
        for (int j = 0; j < 2; ++j) acc[i][j] = wmma4(af[i], bf[j], acc[i][j]);
    }
    async_wait0();
    __syncthreads();
  }

#pragma unroll
  for (int i = 0; i < 2; ++i)
#pragma unroll
    for (int j = 0; j < 2; ++j) {
      const int mBase = row0 + rw0 + i * 16 + ((lane >> 4) << 3);
      const int n = cw0 + j * 16 + (lane & 15);
#pragma unroll
      for (int e = 0; e < 8; ++e)
        Cz[(long long)(mBase + e) * HID + n] = acc[i][j][e];
    }
}

// Row softmax over 2048 columns, in place. One block (8 waves) per row.
__global__ __launch_bounds__(256) void softmax_kernel(float* __restrict__ p) {
  float* row = p + (long long)blockIdx.x * S_LEN;
  const int t = threadIdx.x, wv = t >> 5, ln = t & 31;
  float4* r4 = (float4*)row;
  float4 a = r4[t];
  float4 b = r4[t + 256];

  float m = fmaxf(fmaxf(fmaxf(a.x, a.y), fmaxf(a.z, a.w)),
                  fmaxf(fmaxf(b.x, b.y), fmaxf(b.z, b.w)));
#pragma unroll
  for (int off = 16; off > 0; off >>= 1) m = fmaxf(m, __shfl_xor(m, off, 32));
  __shared__ float redm[8];
  __shared__ float reds[8];
  if (ln == 0) redm[wv] = m;
  __syncthreads();
  m = redm[0];
#pragma unroll
  for (int i = 1; i < 8; ++i) m = fmaxf(m, redm[i]);

  a.x = __expf(a.x - m); a.y = __expf(a.y - m);
  a.z = __expf(a.z - m); a.w = __expf(a.w - m);
  b.x = __expf(b.x - m); b.y = __expf(b.y - m);
  b.z = __expf(b.z - m); b.w = __expf(b.w - m);

  float s = (a.x + a.y + a.z + a.w) + (b.x + b.y + b.z + b.w);
#pragma unroll
  for (int off = 16; off > 0; off >>= 1) s += __shfl_xor(s, off, 32);
  if (ln == 0) reds[wv] = s;
  __syncthreads();
  s = reds[0] + reds[1] + reds[2] + reds[3] + reds[4] + reds[5] + reds[6] + reds[7];
  const float inv = 1.0f / s;

  a.x *= inv; a.y *= inv; a.z *= inv; a.w *= inv;
  b.x *= inv; b.y *= inv; b.z *= inv; b.w *= inv;
  r4[t] = a;
  r4[t + 256] = b;
}

extern "C" void kernel_launch(void* const* d_in, const int* in_sizes, int n_in,
                              void* d_out, int out_size, void* d_ws, size_t ws_size,
                              hipStream_t stream) {
  (void)in_sizes; (void)n_in; (void)out_size; (void)ws_size;
  const float* x  = (const float*)d_in[0];
  const float* Wq = (const float*)d_in[1];
  const float* bq = (const float*)d_in[2];
  const float* Wk = (const float*)d_in[3];
  const float* bk = (const float*)d_in[4];
  const float* Wv = (const float*)d_in[5];
  const float* bv = (const float*)d_in[6];
  const float* Wo = (const float*)d_in[7];
  const float* bo = (const float*)d_in[8];

  const long long BSH = (long long)BATCH * S_LEN * HID;  // 8,388,608
  float* out   = (float*)d_out;
  float* probs = out + BSH;  // output #2 lives directly in d_out
  float* q   = (float*)d_ws;
  float* kb  = q + BSH;
  float* vb  = kb + BSH;
  float* ctx = vb + BSH;

  const dim3 blk(256);
  const dim3 gProj(HID / 128, (BATCH * S_LEN) / 128, 1);  // (8, 64)
  const long long hStride = (long long)S_LEN * HDIM;
  const long long pStride = (long long)S_LEN * S_LEN;

  // Q/K/V projections: pproc forward == x@W.T + b; write head-major [b,h,s,d]
  gemm_bt_kernel<true, false, true><<<gProj, blk, 0, stream>>>(
      x, Wq, bq, nullptr, q, HID, HID, 0, 0, 0);
  gemm_bt_kernel<true, false, true><<<gProj, blk, 0, stream>>>(
      x, Wk, bk, nullptr, kb, HID, HID, 0, 0, 0);
  gemm_bt_kernel<true, false, true><<<gProj, blk, 0, stream>>>(
      x, Wv, bv, nullptr, vb, HID, HID, 0, 0, 0);

  // scores = Q @ K^T per head (no 1/sqrt(d) scaling, per reference)
  gemm_bt_kernel<false, false, false>
      <<<dim3(S_LEN / 128, S_LEN / 128, BATCH * NHEAD), blk, 0, stream>>>(
          q, kb, nullptr, nullptr, probs, HDIM, S_LEN, hStride, hStride, pStride);

  // softmax rows in place (probs is output #2)
  softmax_kernel<<<dim3(BATCH * NHEAD * S_LEN), blk, 0, stream>>>(probs);

  // ctx = probs @ V, written back as [b,s,h*64+d]
  ctx_gemm_kernel<<<dim3(S_LEN / 128, BATCH * NHEAD), blk, 0, stream>>>(probs, vb, ctx);

  // out = ctx @ Wo^T + bo + hidden_states
  gemm_bt_kernel<false, true, true><<<gProj, blk, 0, stream>>>(
      ctx, Wo, bo, x, out, HID, HID, 0, 0, 0);
}